// CostVolumeLoss_377957122733
// MI455X (gfx1250) — compile-verified
//
#include <hip/hip_runtime.h>

#define N_    8
#define C_    3
#define H_    512
#define W_    512
#define K_    5
#define PAD_  2

#define TILE_W 32
#define TILE_H 64
#define HW_    (TILE_W + K_ - 1)          // 36 padded tile width
#define HH_    (TILE_H + K_ - 1)          // 68 padded tile height
#define PLANE_ (HH_ * HW_)                // 2448 floats per channel plane
#define STAGE_TOT (C_ * PLANE_)           // 7344 floats staged per block
#define NBLK_X (W_ / TILE_W)              // 16
#define NBLK_Y (H_ / TILE_H)              // 8
#define NPARTIALS (NBLK_X * NBLK_Y * N_)  // 1024

// -------------------------------------------------------------------------
// Kernel 1: one 32(w) x 64(h) output tile per block (256 thr = 8 waves).
//  - Stage the padded target tile (3 x 68 x 36) into LDS with CDNA5 async
//    global->LDS copies (ASYNCcnt); zero-fill the out-of-image halo.
//  - Overlap: issue pred float4 global loads before waiting on ASYNCcnt.
//  - Each thread owns a 2-row x 4-col pixel patch; the two rows' 5-row
//    windows overlap in 4 rows, so 6 LDS row fetches serve 8 pixels.
// -------------------------------------------------------------------------
extern "C" __global__ __launch_bounds__(256)
void cvl_tile_kernel(const float* __restrict__ pred,
                     const float* __restrict__ target,
                     float* __restrict__ partials)
{
    __shared__ float tile[STAGE_TOT];
    __shared__ float red[256];

    const int tid = threadIdx.x;
    const int n   = blockIdx.z;
    const int h0  = blockIdx.y * TILE_H;
    const int w0  = blockIdx.x * TILE_W;

    // ---- stage padded target tile: async global -> LDS ----
    for (int i = tid; i < STAGE_TOT; i += 256) {
        const int c  = i / PLANE_;
        const int r  = (i % PLANE_) / HW_;
        const int cc = i % HW_;
        const int gr = h0 + r  - PAD_;
        const int gc = w0 + cc - PAD_;
        if ((unsigned)gr < (unsigned)H_ && (unsigned)gc < (unsigned)W_) {
            const float* gp = target + (((size_t)(n * C_ + c) * H_ + gr) * W_ + gc);
            // low 32 bits of a flat LDS pointer == workgroup-relative LDS byte addr
            unsigned ldsoff = (unsigned)(size_t)(&tile[i]);
            asm volatile("global_load_async_to_lds_b32 %0, %1, off"
                         :: "v"(ldsoff), "v"(gp)
                         : "memory");
        } else {
            tile[i] = 0.0f;   // zero padding (nn.Unfold padding semantics)
        }
    }

    // ---- thread's pixel patch: 2 rows x 4 cols ----
    const int rp  = tid >> 3;            // row pair     0..31
    const int cg  = tid & 7;             // col group    0..7
    const int r0  = rp << 1;             // local row    0..62 (step 2)
    const int lc  = cg << 2;             // local col    0,4,...,28
    const int gh0 = h0 + r0;
    const int gw  = w0 + lc;

    // issue pred loads now so they overlap the async staging
    float p[C_][2][4];
#pragma unroll
    for (int c = 0; c < C_; ++c) {
#pragma unroll
        for (int pr = 0; pr < 2; ++pr) {
            const float4 v = *reinterpret_cast<const float4*>(
                pred + (((size_t)(n * C_ + c) * H_ + (gh0 + pr)) * W_ + gw));
            p[c][pr][0] = v.x; p[c][pr][1] = v.y;
            p[c][pr][2] = v.z; p[c][pr][3] = v.w;
        }
    }

#if __has_builtin(__builtin_amdgcn_s_wait_asynccnt)
    __builtin_amdgcn_s_wait_asynccnt(0);
#else
    asm volatile("s_wait_asynccnt 0" ::: "memory");
#endif
    __syncthreads();

    float mn[2][4];
#pragma unroll
    for (int pr = 0; pr < 2; ++pr)
#pragma unroll
        for (int px = 0; px < 4; ++px)
            mn[pr][px] = 3.402823466e38f;

    // roll through 6 padded rows; row rr feeds pixel-row0 (ki=rr, rr<=4)
    // and pixel-row1 (ki=rr-1, rr>=1)
#pragma unroll
    for (int rr = 0; rr < K_ + 1; ++rr) {
        float t[C_][8];
#pragma unroll
        for (int c = 0; c < C_; ++c) {
            // (c*2448 + row*36 + lc)*4 bytes: all terms 16B-aligned
            const float4* rowp = reinterpret_cast<const float4*>(
                &tile[c * PLANE_ + (r0 + rr) * HW_ + lc]);
            const float4 a = rowp[0];
            const float4 b = rowp[1];
            t[c][0] = a.x; t[c][1] = a.y; t[c][2] = a.z; t[c][3] = a.w;
            t[c][4] = b.x; t[c][5] = b.y; t[c][6] = b.z; t[c][7] = b.w;
        }
#pragma unroll
        for (int kj = 0; kj < K_; ++kj) {
#pragma unroll
            for (int px = 0; px < 4; ++px) {
                if (rr <= K_ - 1) {   // pixel row 0
                    const float s = fabsf(p[0][0][px] - t[0][px + kj])
                                  + fabsf(p[1][0][px] - t[1][px + kj])
                                  + fabsf(p[2][0][px] - t[2][px + kj]);
                    mn[0][px] = fminf(mn[0][px], s);
                }
                if (rr >= 1) {        // pixel row 1
                    const float s = fabsf(p[0][1][px] - t[0][px + kj])
                                  + fabsf(p[1][1][px] - t[1][px + kj])
                                  + fabsf(p[2][1][px] - t[2][px + kj]);
                    mn[1][px] = fminf(mn[1][px], s);
                }
            }
        }
    }

    const float part = ((mn[0][0] + mn[0][1]) + (mn[0][2] + mn[0][3]))
                     + ((mn[1][0] + mn[1][1]) + (mn[1][2] + mn[1][3]));

    // ---- deterministic block reduction ----
    red[tid] = part;
    __syncthreads();
#pragma unroll
    for (int off = 128; off > 0; off >>= 1) {
        if (tid < off) red[tid] += red[tid + off];
        __syncthreads();
    }
    if (tid == 0) {
        const int bid = (blockIdx.z * gridDim.y + blockIdx.y) * gridDim.x + blockIdx.x;
        partials[bid] = red[0];
    }
}

// -------------------------------------------------------------------------
// Kernel 2: reduce the 1024 per-block partials into the scalar loss.
// min commutes with the positive 1/C scale, so apply 1/(C*N*H*W) once here.
// -------------------------------------------------------------------------
extern "C" __global__ __launch_bounds__(256)
void cvl_reduce_kernel(const float* __restrict__ partials,
                       float* __restrict__ out)
{
    __shared__ float red[256];
    float s = 0.0f;
    for (int i = threadIdx.x; i < NPARTIALS; i += 256) s += partials[i];
    red[threadIdx.x] = s;
    __syncthreads();
#pragma unroll
    for (int off = 128; off > 0; off >>= 1) {
        if (threadIdx.x < off) red[threadIdx.x] += red[threadIdx.x + off];
        __syncthreads();
    }
    if (threadIdx.x == 0) {
        out[0] = red[0] * (1.0f / (3.0f * (float)N_ * (float)H_ * (float)W_));
    }
}

extern "C" void kernel_launch(void* const* d_in, const int* in_sizes, int n_in,
                              void* d_out, int out_size, void* d_ws, size_t ws_size,
                              hipStream_t stream)
{
    const float* pred   = (const float*)d_in[0];
    const float* target = (const float*)d_in[1];
    float* partials     = (float*)d_ws;   // 1024 floats = 4 KB scratch

    dim3 grid(NBLK_X, NBLK_Y, N_);
    cvl_tile_kernel<<<grid, 256, 0, stream>>>(pred, target, partials);
    cvl_reduce_kernel<<<1, 256, 0, stream>>>(partials, (float*)d_out);
}